// SimpleGNN_76184129896627
// MI455X (gfx1250) — compile-verified
//
#include <hip/hip_runtime.h>

#define NNODES 8192
#define NEDGES 262144
#define INDIM  128
#define HDIM   256

typedef float v2f __attribute__((ext_vector_type(2)));
typedef float v4f __attribute__((ext_vector_type(4)));
typedef float v8f __attribute__((ext_vector_type(8)));

// ---------------- degree / normalization ----------------
__global__ __launch_bounds__(256) void deg_init(int* deg) {
    int i = blockIdx.x * 256 + threadIdx.x;
    deg[i] = 1;  // self loop
}

__global__ __launch_bounds__(256) void deg_count(const long long* __restrict__ ei, int* deg) {
    int e = blockIdx.x * 256 + threadIdx.x;
    long long d = ei[NEDGES + e];             // dst row of edge_index
    atomicAdd(&deg[d], 1);
}

__global__ __launch_bounds__(256) void dinv_finalize(const int* __restrict__ cnt, float* dinv) {
    int i = blockIdx.x * 256 + threadIdx.x;
    float c = (float)cnt[i];
    dinv[i] = rsqrtf(c);                      // deg always >= 1 (self loop)
}

// ---------------- f32 WMMA GEMM: Y[M,N] = act(X)[M,K] @ W[N,K]^T (+bias) ----------------
// One wave computes one 16x16 tile with V_WMMA_F32_16X16X4_F32.
// RELU/BIAS are compile-time so the K-loop is pure load + (max) + wmma.
template <bool RELU, bool BIAS>
__global__ __launch_bounds__(256) void gemm_wmma(const float* __restrict__ X,
                                                 const float* __restrict__ W,
                                                 const float* __restrict__ bias,
                                                 float* __restrict__ Y,
                                                 int M, int K, int N) {
    int lane = threadIdx.x & 31;
    int wid  = threadIdx.x >> 5;
    int tile = blockIdx.x * 8 + wid;
    int ntn  = N >> 4;
    int mt   = tile / ntn;
    int nt   = tile - mt * ntn;

    int lr = lane & 15;           // row-in-tile (A) / col-in-tile (B)
    int kh = (lane >> 4) << 1;    // lanes 0-15: K offset 0,1 ; lanes 16-31: K offset 2,3

    const float* xr = X + (size_t)(mt * 16 + lr) * K;
    const float* wr = W + (size_t)(nt * 16 + lr) * K;

    v8f acc = {};
    for (int k = 0; k < K; k += 4) {
        v2f a, b;
        a[0] = ((const v2f*)(xr + k + kh))[0][0];
        a[1] = ((const v2f*)(xr + k + kh))[0][1];
        if (RELU) { a[0] = fmaxf(a[0], 0.0f); a[1] = fmaxf(a[1], 0.0f); }
        b[0] = ((const v2f*)(wr + k + kh))[0][0];
        b[1] = ((const v2f*)(wr + k + kh))[0][1];
        acc = __builtin_amdgcn_wmma_f32_16x16x4_f32(
            /*neg_a=*/false, a, /*neg_b=*/false, b,
            /*c_mod=*/(short)0, acc, /*reuse_a=*/false, /*reuse_b=*/false);
    }

    int n     = nt * 16 + lr;
    int mbase = mt * 16 + ((lane >> 4) << 3);   // lanes 0-15: rows 0..7 ; 16-31: rows 8..15
    float bv  = BIAS ? bias[n] : 0.0f;
    #pragma unroll
    for (int r = 0; r < 8; ++r)
        Y[(size_t)(mbase + r) * N + n] = acc[r] + bv;
}

// ---------------- agg init: agg = bias + t * dinv^2  (self-loop term + bias) ----------------
__global__ __launch_bounds__(256) void init_agg(const float* __restrict__ t,
                                                const float* __restrict__ dinv,
                                                const float* __restrict__ bias,
                                                float* __restrict__ agg) {
    size_t idx = (size_t)blockIdx.x * 256 + threadIdx.x;   // float4 index
    const int Hq = HDIM >> 2;                              // 64
    int m  = (int)(idx / Hq);
    int j4 = (int)(idx - (size_t)m * Hq);
    float d  = dinv[m];
    float dd = d * d;
    v4f tv = ((const v4f*)t)[idx];
    v4f bv = ((const v4f*)bias)[j4];
    v4f r  = { bv[0] + tv[0] * dd, bv[1] + tv[1] * dd,
               bv[2] + tv[2] * dd, bv[3] + tv[3] * dd };
    ((v4f*)agg)[idx] = r;
}

// ---------------- edge scatter: agg[dst] += t[src] * dinv[src]*dinv[dst] ----------------
// One wave per edge; 256 features / 32 lanes = 8 atomics per lane (L2-resident).
__global__ __launch_bounds__(256) void scatter_edges(const long long* __restrict__ ei,
                                                     const float* __restrict__ t,
                                                     const float* __restrict__ dinv,
                                                     float* __restrict__ agg) {
    int lane = threadIdx.x & 31;
    int wid  = threadIdx.x >> 5;
    int e    = blockIdx.x * 8 + wid;
    long long s = ei[e];
    long long d = ei[NEDGES + e];
    float norm = dinv[s] * dinv[d];
    const float* ts = t   + (size_t)s * HDIM;
    float*       ad = agg + (size_t)d * HDIM;
    #pragma unroll
    for (int j = 0; j < HDIM / 32; ++j) {
        int col = lane + j * 32;
        unsafeAtomicAdd(&ad[col], ts[col] * norm);
    }
}

// ---------------- per-node dots: zi = h@wi, zj = h@wj ----------------
__global__ __launch_bounds__(256) void node_dots(const float* __restrict__ h,
                                                 const float* __restrict__ ew,
                                                 float* __restrict__ zi,
                                                 float* __restrict__ zj) {
    int lane = threadIdx.x & 31;
    int wid  = threadIdx.x >> 5;
    int node = blockIdx.x * 8 + wid;
    const float* hp = h + (size_t)node * HDIM;
    float si = 0.0f, sj = 0.0f;
    #pragma unroll
    for (int j = 0; j < HDIM / 32; ++j) {
        int col = lane + j * 32;
        float hv = hp[col];
        si += hv * ew[col];
        sj += hv * ew[HDIM + col];
    }
    #pragma unroll
    for (int m = 16; m >= 1; m >>= 1) {
        si += __shfl_xor(si, m, 32);
        sj += __shfl_xor(sj, m, 32);
    }
    if (lane == 0) { zi[node] = si; zj[node] = sj; }
}

// ---------------- logits[i,j] = zi[i] + zj[j] + eb  (256MB NT stream-out) ----------------
__global__ __launch_bounds__(256) void edge_logits(const float* __restrict__ zi,
                                                   const float* __restrict__ zj,
                                                   const float* __restrict__ ebp,
                                                   float* __restrict__ out) {
    int j4 = blockIdx.x * 256 + threadIdx.x;     // float4 column index
    int i  = blockIdx.y;
    float base = zi[i] + ebp[0];
    v4f v = ((const v4f*)zj)[j4];
    v4f r = { v[0] + base, v[1] + base, v[2] + base, v[3] + base };
    __builtin_nontemporal_store(r, (v4f*)out + (size_t)i * (NNODES / 4) + j4);
}

extern "C" void kernel_launch(void* const* d_in, const int* in_sizes, int n_in,
                              void* d_out, int out_size, void* d_ws, size_t ws_size,
                              hipStream_t stream) {
    const float*     x    = (const float*)d_in[0];
    const long long* ei   = (const long long*)d_in[1];   // int64 (2, E)
    const float*     W0   = (const float*)d_in[2];
    const float*     b0   = (const float*)d_in[3];
    const float*     W1   = (const float*)d_in[4];
    const float*     b1   = (const float*)d_in[5];
    const float*     W2   = (const float*)d_in[6];
    const float*     b2   = (const float*)d_in[7];
    const float*     outW = (const float*)d_in[8];
    const float*     outb = (const float*)d_in[9];
    const float*     edgW = (const float*)d_in[10];      // (1, 512): wi | wj
    const float*     edgb = (const float*)d_in[11];      // (1,)

    float* buf0 = (float*)d_ws;                          // 8192*256 f32 = 8 MB
    float* buf1 = buf0 + (size_t)NNODES * HDIM;          // 8 MB
    float* dinv = buf1 + (size_t)NNODES * HDIM;          // 32 KB (aliased as int deg)
    float* zi   = dinv + NNODES;
    float* zj   = zi + NNODES;
    int*   deg  = (int*)dinv;

    // 1) symmetric normalization dinv = (indeg + 1)^(-1/2)
    deg_init<<<NNODES / 256, 256, 0, stream>>>(deg);
    deg_count<<<NEDGES / 256, 256, 0, stream>>>(ei, deg);
    dinv_finalize<<<NNODES / 256, 256, 0, stream>>>(deg, dinv);

    const int tilesBlk = (NNODES / 16) * (HDIM / 16) / 8;  // gemm blocks (8 waves each)
    const int aggBlk   = (NNODES * HDIM / 4) / 256;
    const int edgBlk   = NEDGES / 8;

    // 2) layer 0: t = x @ W0^T ; agg = b0 + t*dinv^2 ; scatter
    gemm_wmma<false, false><<<tilesBlk, 256, 0, stream>>>(x, W0, nullptr, buf0, NNODES, INDIM, HDIM);
    init_agg <<<aggBlk, 256, 0, stream>>>(buf0, dinv, b0, buf1);
    scatter_edges<<<edgBlk, 256, 0, stream>>>(ei, buf0, dinv, buf1);

    // 3) layer 1: t = relu(agg) @ W1^T ; agg = b1 + t*dinv^2 ; scatter
    gemm_wmma<true, false><<<tilesBlk, 256, 0, stream>>>(buf1, W1, nullptr, buf0, NNODES, HDIM, HDIM);
    init_agg <<<aggBlk, 256, 0, stream>>>(buf0, dinv, b1, buf1);
    scatter_edges<<<edgBlk, 256, 0, stream>>>(ei, buf0, dinv, buf1);

    // 4) layer 2
    gemm_wmma<true, false><<<tilesBlk, 256, 0, stream>>>(buf1, W2, nullptr, buf0, NNODES, HDIM, HDIM);
    init_agg <<<aggBlk, 256, 0, stream>>>(buf0, dinv, b2, buf1);
    scatter_edges<<<edgBlk, 256, 0, stream>>>(ei, buf0, dinv, buf1);

    // 5) output projection: h = relu(agg) @ outW^T + outb
    gemm_wmma<true, true><<<tilesBlk, 256, 0, stream>>>(buf1, outW, outb, buf0, NNODES, HDIM, HDIM);

    // 6) edge-scoring rank-1 logits
    node_dots<<<NNODES / 8, 256, 0, stream>>>(buf0, edgW, zi, zj);
    dim3 lg(NNODES / 4 / 256, NNODES);
    edge_logits<<<lg, 256, 0, stream>>>(zi, zj, edgb, (float*)d_out);
}